// BilinearSampler_50800873177201
// MI455X (gfx1250) — compile-verified
//
#include <hip/hip_runtime.h>

// BilinearSampler (spatial-transformer sampler), B=128, H=W=256, C=3, fp32.
// Memory-bound gather: ~200 MB traffic -> ~8.6 us floor at 23.3 TB/s.
// No matrix op of meaningful shape exists here (theta is 2x3, K=3), so no
// WMMA path; the optimal mix is scalar theta loads + b96 vector gathers.

#define BB 128
#define HH 256
#define WW 256
#define CC 3
#define HW (HH * WW)

struct __align__(4) f3 { float x, y, z; };

__global__ __launch_bounds__(256) void bilinear_sampler_kernel(
    const float* __restrict__ in,   // (B, 6 + H*W*C)
    float* __restrict__ out)        // (B, H, W, C)
{
    const int b = blockIdx.y;
    const int p = blockIdx.x * blockDim.x + threadIdx.x;  // pixel index in H*W

    // Per-batch base: uniform across the block -> theta loads scalarize (SMEM).
    const float* __restrict__ base = in + (size_t)b * (size_t)(6 + HW * CC);
    const float t00 = base[0], t01 = base[1], t02 = base[2];
    const float t10 = base[3], t11 = base[4], t12 = base[5];
    const float* __restrict__ img = base + 6;            // (H, W, C)

    // Decompose pixel (W = 256 -> shift/mask).
    const int h = p >> 8;
    const int w = p & (WW - 1);

    // Normalized grid coords in [-1, 1] (linspace over W-1 / H-1 steps).
    const float xt = fmaf((float)w, 2.0f / (float)(WW - 1), -1.0f);
    const float yt = fmaf((float)h, 2.0f / (float)(HH - 1), -1.0f);

    // Affine transform (the "einsum"): 4 FMAs.
    const float xs = fmaf(t00, xt, fmaf(t01, yt, t02));
    const float ys = fmaf(t10, xt, fmaf(t11, yt, t12));

    // Map to pixel space.
    const float x = 0.5f * (xs + 1.0f) * (float)(WW - 1);
    const float y = 0.5f * (ys + 1.0f) * (float)(HH - 1);

    const int x0 = (int)floorf(x);
    const int y0 = (int)floorf(y);
    const int x1 = x0 + 1;
    const int y1 = y0 + 1;

    // Clamp (reference clamps BEFORE computing the float corner values).
    const int x0c = min(max(x0, 0), WW - 1);
    const int x1c = min(max(x1, 0), WW - 1);
    const int y0c = min(max(y0, 0), HH - 1);
    const int y1c = min(max(y1, 0), HH - 1);

    const float x0f = (float)x0c;
    const float x1f = (float)x1c;
    const float y0f = (float)y0c;
    const float y1f = (float)y1c;

    const float wa = (x1f - x) * (y1f - y);
    const float wb = (x1f - x) * (y - y0f);
    const float wc = (x - x0f) * (y1f - y);
    const float wd = (x - x0f) * (y - y0f);

    // Gather the 4 corner pixels (12 contiguous bytes each -> b96 loads).
    const f3* __restrict__ pix = (const f3*)img;
    const f3 Ia = pix[y0c * WW + x0c];
    const f3 Ib = pix[y1c * WW + x0c];
    const f3 Ic = pix[y0c * WW + x1c];
    const f3 Id = pix[y1c * WW + x1c];

    f3 r;
    r.x = fmaf(wa, Ia.x, fmaf(wb, Ib.x, fmaf(wc, Ic.x, wd * Id.x)));
    r.y = fmaf(wa, Ia.y, fmaf(wb, Ib.y, fmaf(wc, Ic.y, wd * Id.y)));
    r.z = fmaf(wa, Ia.z, fmaf(wb, Ib.z, fmaf(wc, Ic.z, wd * Id.z)));

    // Coalesced b96 store: wave32 writes 384 contiguous bytes.
    ((f3*)out)[(size_t)b * HW + p] = r;
}

extern "C" void kernel_launch(void* const* d_in, const int* in_sizes, int n_in,
                              void* d_out, int out_size, void* d_ws, size_t ws_size,
                              hipStream_t stream) {
    (void)n_in; (void)out_size; (void)d_ws; (void)ws_size;
    const float* in = (const float*)d_in[0];
    float* out = (float*)d_out;

    // Batch count from input size (robust to harness changes); H/W/C fixed.
    const int batches = in_sizes[0] / (6 + HW * CC);

    dim3 block(256, 1, 1);
    dim3 grid(HW / 256, batches, 1);
    bilinear_sampler_kernel<<<grid, block, 0, stream>>>(in, out);
}